// ChamferDistance_29910152250052
// MI455X (gfx1250) — compile-verified
//
#include <hip/hip_runtime.h>

typedef __attribute__((ext_vector_type(2))) float v2f;
typedef __attribute__((ext_vector_type(8))) float v8f;

#define WAVES_PER_BLOCK 4
#define ROWS_PER_WAVE   16
#define ROWS_PER_BLOCK  (WAVES_PER_BLOCK * ROWS_PER_WAVE)

// ---------------------------------------------------------------------------
// Kernel 0: squared norms of every point (3 components each).
// ---------------------------------------------------------------------------
__global__ __launch_bounds__(256)
void chamfer_sqnorm_kernel(const float* __restrict__ pts,
                           float* __restrict__ norms,
                           int total)
{
    int i = blockIdx.x * blockDim.x + threadIdx.x;
    if (i < total) {
        float x = pts[3 * i + 0];
        float y = pts[3 * i + 1];
        float z = pts[3 * i + 2];
        norms[i] = x * x + y * y + z * z;
    }
}

// ---------------------------------------------------------------------------
// One chamfer direction: for each query point in P (per batch), find
// min / argmin over all reference points in Q of ||p - q||^2.
//
// ||p-q||^2 = ||p||^2 + (||q||^2 - 2 p.q). The parenthesized key is what we
// minimize in the loop; ||p||^2 is added back at the end (row-constant).
//
// The 2*p.q cross term is computed 16x16 at a time with
// V_WMMA_F32_16X16X4_F32 (C=3 padded to K=4 with zeros).
//
// A (16x4, MxK) per-lane layout:   row M = lane%16,
//   VGPR0 = K(2*half), VGPR1 = K(2*half+1), half = lane/16. K=3 is zero pad.
// B (4x16, KxN) per-lane layout:   col N = lane%16, mirrored K slots.
// D (16x16 f32) layout: VGPR v of lane l -> element (M = v + 8*(l/16),
//   N = l%16), so each lane tracks 8 row-minima for its column slice.
// ---------------------------------------------------------------------------
__global__ __launch_bounds__(WAVES_PER_BLOCK * 32)
void chamfer_pass_kernel(const float* __restrict__ P,     // [B, Np, 3] queries
                         const float* __restrict__ Q,     // [B, Nq, 3] references
                         const float* __restrict__ Pn,    // [B*Np] ||p||^2
                         const float* __restrict__ Qn,    // [B*Nq] ||q||^2
                         float* __restrict__ minval,      // [B*Np] min squared dist
                         float* __restrict__ idx_out,     // [B*Np] argmin, value-cast
                         int Np, int Nq)
{
    const int lane = threadIdx.x & 31;
    const int wave = threadIdx.x >> 5;
    const int half = lane >> 4;       // 0: lanes 0-15, 1: lanes 16-31
    const int l16  = lane & 15;

    const int tilesPerBatch = Np / ROWS_PER_BLOCK;
    const int b  = blockIdx.x / tilesPerBatch;
    const int n0 = (blockIdx.x % tilesPerBatch) * ROWS_PER_BLOCK
                 + wave * ROWS_PER_WAVE;

    const float* Pb  = P  + (size_t)b * Np * 3;
    const float* Qb  = Q  + (size_t)b * Nq * 3;
    const float* Qnb = Qn + (size_t)b * Nq;

    // --- A operand: this wave's 16 query rows, loaded once, kept in VGPRs ---
    const int arow = n0 + l16;
    float ax = Pb[3 * arow + 2 * half];     // comp 0 (half==0) or comp 2 (half==1)
    float ay = Pb[3 * arow + 1];            // load full-exec, mask below
    v2f a;
    a.x = ax;
    a.y = half ? 0.0f : ay;                 // K=3 zero pad for upper half

    // Running (min, argmin) of key = ||q||^2 - 2 p.q, 8 rows per lane.
    float best[8];
    int   besti[8];
#pragma unroll
    for (int v = 0; v < 8; ++v) { best[v] = 3.4e38f; besti[v] = 0; }

    // --- sweep all reference tiles (16 points each) ---
    for (int m0 = 0; m0 < Nq; m0 += 16) {
        const int col = m0 + l16;
        float bx = Qb[3 * col + 2 * half];
        float by = Qb[3 * col + 1];
        v2f bb;
        bb.x = bx;
        bb.y = half ? 0.0f : by;
        const float qn = Qnb[col];          // ||q_col||^2, per-lane column

        v8f c = {};
        c = __builtin_amdgcn_wmma_f32_16x16x4_f32(
                /*neg_a=*/false, a, /*neg_b=*/false, bb,
                /*c_mod=*/(short)0, c, /*reuse_a=*/false, /*reuse_b=*/false);

#pragma unroll
        for (int v = 0; v < 8; ++v) {
            float key = qn - 2.0f * c[v];
            if (key < best[v]) { best[v] = key; besti[v] = col; }
        }
    }

    // --- cross-lane min/argmin within each 16-lane half (wave32 shuffles) ---
#pragma unroll
    for (int v = 0; v < 8; ++v) {
        float bv = best[v];
        int   bi = besti[v];
#pragma unroll
        for (int off = 8; off >= 1; off >>= 1) {
            float ov = __shfl_xor(bv, off, 32);
            int   oi = __shfl_xor(bi, off, 32);
            if (ov < bv || (ov == bv && oi < bi)) { bv = ov; bi = oi; }
        }
        if (l16 == 0) {
            const int row = n0 + v + 8 * half;
            const size_t gi = (size_t)b * Np + row;
            minval[gi]  = bv + Pn[gi];      // add back ||p||^2
            idx_out[gi] = (float)bi;        // argmin, value-cast to f32
        }
    }
}

// ---------------------------------------------------------------------------
// Final reduction: means of the two min-distance arrays (single block).
// ---------------------------------------------------------------------------
__global__ __launch_bounds__(256)
void chamfer_finalize_kernel(const float* __restrict__ mv_src, int n_src,
                             const float* __restrict__ mv_dst, int n_dst,
                             float* __restrict__ out)
{
    __shared__ float s1[256];
    __shared__ float s2[256];
    float acc1 = 0.0f, acc2 = 0.0f;
    for (int i = threadIdx.x; i < n_src; i += 256) acc1 += mv_src[i];
    for (int i = threadIdx.x; i < n_dst; i += 256) acc2 += mv_dst[i];
    s1[threadIdx.x] = acc1;
    s2[threadIdx.x] = acc2;
    __syncthreads();
    for (int s = 128; s > 0; s >>= 1) {
        if (threadIdx.x < s) {
            s1[threadIdx.x] += s1[threadIdx.x + s];
            s2[threadIdx.x] += s2[threadIdx.x + s];
        }
        __syncthreads();
    }
    if (threadIdx.x == 0) {
        out[0] = s1[0] / (float)n_src;   // loss_src (mean, weight 1.0)
        out[1] = s2[0] / (float)n_dst;   // loss_dst (mean, weight 1.0)
    }
}

// ---------------------------------------------------------------------------
// Launch wrapper.
// d_out layout (float): [loss_src, loss_dst, indices1 (B*N), indices2 (B*M)]
// d_ws layout (float):  [src_norm B*N | tgt_norm B*M | mv_src B*N | mv_dst B*M]
// ---------------------------------------------------------------------------
extern "C" void kernel_launch(void* const* d_in, const int* in_sizes, int n_in,
                              void* d_out, int out_size, void* d_ws, size_t ws_size,
                              hipStream_t stream)
{
    const int B = 2;
    const int N = in_sizes[0] / (B * 3);   // 8192
    const int M = in_sizes[1] / (B * 3);   // 8192

    const float* src = (const float*)d_in[0];   // [B, N, 3]
    const float* tgt = (const float*)d_in[1];   // [B, M, 3]
    float* out = (float*)d_out;
    float* ws  = (float*)d_ws;

    float* src_norm = ws;                       // B*N
    float* tgt_norm = src_norm + B * N;         // B*M
    float* mv_src   = tgt_norm + B * M;         // B*N
    float* mv_dst   = mv_src   + B * N;         // B*M
    float* idx1     = out + 2;                  // B*N (indices1 as float)
    float* idx2     = idx1 + B * N;             // B*M (indices2 as float)

    chamfer_sqnorm_kernel<<<(B * N + 255) / 256, 256, 0, stream>>>(src, src_norm, B * N);
    chamfer_sqnorm_kernel<<<(B * M + 255) / 256, 256, 0, stream>>>(tgt, tgt_norm, B * M);

    dim3 block(WAVES_PER_BLOCK * 32);
    // src -> tgt : per source point, nearest target
    chamfer_pass_kernel<<<B * (N / ROWS_PER_BLOCK), block, 0, stream>>>(
        src, tgt, src_norm, tgt_norm, mv_src, idx1, N, M);
    // tgt -> src : per target point, nearest source
    chamfer_pass_kernel<<<B * (M / ROWS_PER_BLOCK), block, 0, stream>>>(
        tgt, src, tgt_norm, src_norm, mv_dst, idx2, M, N);

    chamfer_finalize_kernel<<<1, 256, 0, stream>>>(mv_src, B * N, mv_dst, B * M, out);
}